// MultiHeadAdditiveAttention_7619271983257
// MI455X (gfx1250) — compile-verified
//
#include <hip/hip_runtime.h>
#include <hip/hip_bf16.h>

// Problem constants (match reference)
#define B_  4
#define LQ_ 128
#define LK_ 256
#define D_  128
#define H_  128
#define NH_ 8

typedef float v2f __attribute__((ext_vector_type(2)));
typedef float v8f __attribute__((ext_vector_type(8)));

// ---------------------------------------------------------------------------
// Wave-level (wave32) fp32 GEMM tile: C(16x16) = sum_k A[16xK] * B[KxN]
// A is row-major with leading dim lda (contiguous in k).
// B element (k,n) is at Bp[k*sbk + n*sbn]  (generic strides => free transpose).
// Uses V_WMMA_F32_16X16X4_F32; fragment layouts per CDNA5 ISA 7.12.2:
//   A frag (2 VGPR):  a[j] = A[l][k0 + 2*half + j]
//   B frag (2 VGPR):  b[j] = B[k0 + 2*half + j][l]
//   C/D   (8 VGPR):   c[r] = C[r + 8*half][l]
// ---------------------------------------------------------------------------
__device__ __forceinline__ v8f wmma_gemm16(const float* __restrict__ A, int lda,
                                           const float* __restrict__ Bp, int sbk, int sbn,
                                           int K, int lane) {
  const int l   = lane & 15;
  const int kh2 = (lane >> 4) << 1;   // 0 or 2
  const float* arow = A + l * lda;
  const float* bcol = Bp + l * sbn;
  v8f c = {};
  for (int k0 = 0; k0 < K; k0 += 4) {
    v2f a, b;
    a.x = arow[k0 + kh2 + 0];
    a.y = arow[k0 + kh2 + 1];
    b.x = bcol[(k0 + kh2 + 0) * sbk];
    b.y = bcol[(k0 + kh2 + 1) * sbk];
    c = __builtin_amdgcn_wmma_f32_16x16x4_f32(false, a, false, b, (short)0, c, false, false);
  }
  return c;
}

// Native V_TANH_F32 when the builtin exists (gfx1250 trans-unit op); otherwise
// synthesize from v_exp_f32 + v_rcp_f32 (two quarter-rate trans ops).
__device__ __forceinline__ float fast_tanh(float x) {
#if __has_builtin(__builtin_amdgcn_tanhf)
  return __builtin_amdgcn_tanhf(x);
#elif __has_builtin(__builtin_amdgcn_tanh_f32)
  return __builtin_amdgcn_tanh_f32(x);
#else
  // tanh(x) = 1 - 2/(exp(2x)+1); exp(2x) = exp2(x * 2*log2(e))
  float e = __builtin_amdgcn_exp2f(x * 2.8853900817779268f);
  return 1.0f - 2.0f * __builtin_amdgcn_rcpf(e + 1.0f);
#endif
}

// ---------------------------------------------------------------------------
// Kernel 1: per-head projection  Y[b,n,m,h] = sum_d X[b,m,d] * W[n,h,d]
// One wave (32 threads) per 16x16 output tile.
// ---------------------------------------------------------------------------
__global__ void proj_kernel(const float* __restrict__ X, const float* __restrict__ W,
                            float* __restrict__ Y, int L) {
  const int tilesN = H_ / 16;            // 8
  const int tilesM = L / 16;
  const int per_bn = tilesM * tilesN;
  int t  = blockIdx.x;
  int bn = t / per_bn;
  int tt = t % per_bn;
  int m0 = (tt / tilesN) * 16;
  int n0 = (tt % tilesN) * 16;
  int b  = bn / NH_;
  int n  = bn % NH_;
  int lane = threadIdx.x & 31;

  const float* A  = X + (b * L + m0) * D_;             // lda = D
  const float* Bp = W + (n * H_ + n0) * D_;            // B(k,h)=W[n][n0+h][k]: sbk=1, sbn=D
  v8f c = wmma_gemm16(A, D_, Bp, 1, D_, D_, lane);

  float* C = Y + ((b * NH_ + n) * L + m0) * H_ + n0;
  int l = lane & 15, mh = (lane >> 4) << 3;
#pragma unroll
  for (int r = 0; r < 8; ++r) C[(mh + r) * H_ + l] = c[r];
}

// ---------------------------------------------------------------------------
// Kernel 2: additive scores + masked softmax (the transcendental hot loop).
// One 256-thread block per (b, n, 8-q block). Each wave owns one q row;
// each lane owns k = lane + 32*c (c = 0..7). kh tiled through LDS in 64-row
// chunks. Row stride H+4 floats (528 B): 16-byte aligned so each lane reads
// kh with ds_load_b128 (512 B/wave = LDS bandwidth floor; worst 2-way bank
// rotation). qh / wv reads are same-address broadcasts.
// ---------------------------------------------------------------------------
#define QT 8
#define KCH 64
#define PAD 4
__global__ __launch_bounds__(256) void scores_softmax_kernel(
    const float* __restrict__ qh, const float* __restrict__ kh,
    const float* __restrict__ wv, const int* __restrict__ valid_lens,
    float* __restrict__ attn)
{
  const int qblocks = LQ_ / QT;          // 16
  int t  = blockIdx.x;
  int qb = t % qblocks; t /= qblocks;
  int n  = t % NH_;
  int b  = t / NH_;
  const int tid  = threadIdx.x;
  const int lane = tid & 31;
  const int ql   = tid >> 5;             // wave id == local q row

  __shared__ __align__(16) float s_qh[QT][H_ + PAD];
  __shared__ __align__(16) float s_wv[H_];
  __shared__ __align__(16) float s_kh[KCH][H_ + PAD];

  const float* qhp = qh + ((b * NH_ + n) * LQ_ + qb * QT) * H_;
  for (int i = tid; i < QT * H_; i += 256) s_qh[i / H_][i % H_] = qhp[i];
  for (int i = tid; i < H_; i += 256)      s_wv[i] = wv[n * H_ + i];

  const float* khp = kh + (b * NH_ + n) * LK_ * H_;
  const float4* qr = (const float4*)&s_qh[ql][0];
  const float4* wr = (const float4*)&s_wv[0];

  float sc[8];
  for (int kc = 0; kc < LK_ / KCH; ++kc) {
    __syncthreads();
    for (int i = tid; i < KCH * H_; i += 256)
      s_kh[i / H_][i % H_] = khp[kc * KCH * H_ + i];
    __syncthreads();
#pragma unroll
    for (int j = 0; j < 2; ++j) {
      const int kl = lane + 32 * j;
      const float4* kr = (const float4*)&s_kh[kl][0];
      float s = 0.0f;
#pragma unroll 4
      for (int h4 = 0; h4 < H_ / 4; ++h4) {
        float4 q = qr[h4];
        float4 k = kr[h4];
        float4 w = wr[h4];
        s += w.x * fast_tanh(q.x + k.x);
        s += w.y * fast_tanh(q.y + k.y);
        s += w.z * fast_tanh(q.z + k.z);
        s += w.w * fast_tanh(q.w + k.w);
      }
      sc[kc * 2 + j] = s;                // global k = 32*(kc*2+j) + lane
    }
  }

  // mask (reference: scores := -1e6 where k >= valid_len, then exact softmax)
  const int vlen = valid_lens[b];
#pragma unroll
  for (int c = 0; c < 8; ++c) {
    int k = lane + 32 * c;
    if (k >= vlen) sc[c] = -1.0e6f;
  }

  // softmax over 256 values: 8 regs/lane + wave32 reduction
  float m = sc[0];
#pragma unroll
  for (int c = 1; c < 8; ++c) m = fmaxf(m, sc[c]);
  for (int off = 16; off >= 1; off >>= 1) m = fmaxf(m, __shfl_xor(m, off, 32));

  float e[8];
  float sum = 0.0f;
#pragma unroll
  for (int c = 0; c < 8; ++c) {
    e[c] = __builtin_amdgcn_exp2f((sc[c] - m) * 1.4426950408889634f);
    sum += e[c];
  }
  for (int off = 16; off >= 1; off >>= 1) sum += __shfl_xor(sum, off, 32);
  float r = __builtin_amdgcn_rcpf(sum);

  float* ap = attn + ((b * NH_ + n) * LQ_ + qb * QT + ql) * LK_;
#pragma unroll
  for (int c = 0; c < 8; ++c) ap[32 * c + lane] = e[c] * r;
}

// ---------------------------------------------------------------------------
// Kernel 3: out_heads = attn @ V, stored directly in concat-head layout
// concat[b, q, n*D + v]. One wave per 16x16 tile, K = LK.
// ---------------------------------------------------------------------------
__global__ void av_kernel(const float* __restrict__ attn, const float* __restrict__ V,
                          float* __restrict__ concat) {
  int t  = blockIdx.x;
  int bn = t / 64;                       // (LQ/16)*(D/16)=64 tiles per (b,n)
  int tt = t % 64;
  int m0 = (tt / 8) * 16;
  int n0 = (tt % 8) * 16;
  int b  = bn / NH_;
  int n  = bn % NH_;
  int lane = threadIdx.x & 31;

  const float* A  = attn + (bn * LQ_ + m0) * LK_;      // lda = LK
  const float* Bp = V + b * LK_ * D_ + n0;             // B(k,v)=V[b][k][n0+v]: sbk=D, sbn=1
  v8f c = wmma_gemm16(A, LK_, Bp, D_, 1, LK_, lane);

  float* C = concat + (b * LQ_ + m0) * (NH_ * D_) + n * D_ + n0;
  int l = lane & 15, mh = (lane >> 4) << 3;
#pragma unroll
  for (int r = 0; r < 8; ++r) C[(mh + r) * (NH_ * D_) + l] = c[r];
}

// ---------------------------------------------------------------------------
// Kernel 4: out = concat @ Wo_w^T + Wo_b.  M = B*LQ = 512, N = H, K = NH*D.
// ---------------------------------------------------------------------------
__global__ void out_kernel(const float* __restrict__ concat, const float* __restrict__ Wo,
                           const float* __restrict__ bo, float* __restrict__ out) {
  const int K = NH_ * D_;                // 1024
  int t  = blockIdx.x;
  int m0 = (t / 8) * 16;                 // 32 m-tiles
  int n0 = (t % 8) * 16;
  int lane = threadIdx.x & 31;

  const float* A  = concat + m0 * K;                   // lda = K
  const float* Bp = Wo + n0 * K;                       // B(k,h)=Wo[n0+h][k]: sbk=1, sbn=K
  v8f c = wmma_gemm16(A, K, Bp, 1, K, K, lane);

  int l = lane & 15, mh = (lane >> 4) << 3;
  float bias = bo[n0 + l];
#pragma unroll
  for (int r = 0; r < 8; ++r) out[(m0 + mh + r) * H_ + n0 + l] = c[r] + bias;
}

// ---------------------------------------------------------------------------
extern "C" void kernel_launch(void* const* d_in, const int* in_sizes, int n_in,
                              void* d_out, int out_size, void* d_ws, size_t ws_size,
                              hipStream_t stream) {
  const float* queries = (const float*)d_in[0];
  const float* keys    = (const float*)d_in[1];
  const float* values  = (const float*)d_in[2];
  const int*   vlens   = (const int*)  d_in[3];
  const float* Wq      = (const float*)d_in[4];
  const float* Wk      = (const float*)d_in[5];
  const float* wv      = (const float*)d_in[6];
  const float* Wo      = (const float*)d_in[7];
  const float* bo      = (const float*)d_in[8];
  float* out = (float*)d_out;

  // workspace carving (floats): qh | kh | attn | concat  (~12 MB total)
  float* qh     = (float*)d_ws;
  float* kh     = qh   + (size_t)B_ * NH_ * LQ_ * H_;
  float* attn   = kh   + (size_t)B_ * NH_ * LK_ * H_;
  float* concat = attn + (size_t)B_ * NH_ * LQ_ * LK_;

  // 1) per-head projections (fp32 WMMA)
  proj_kernel<<<B_ * NH_ * (LQ_ / 16) * (H_ / 16), 32, 0, stream>>>(queries, Wq, qh, LQ_);
  proj_kernel<<<B_ * NH_ * (LK_ / 16) * (H_ / 16), 32, 0, stream>>>(keys,    Wk, kh, LK_);

  // 2) additive scores + masked softmax (trans-unit bound hot loop)
  scores_softmax_kernel<<<B_ * NH_ * (LQ_ / QT), 256, 0, stream>>>(qh, kh, wv, vlens, attn);

  // 3) attn @ V into concat-head layout (fp32 WMMA)
  av_kernel<<<B_ * NH_ * 64, 32, 0, stream>>>(attn, values, concat);

  // 4) output projection + bias (fp32 WMMA)
  out_kernel<<<(B_ * LQ_ / 16) * (H_ / 16), 32, 0, stream>>>(concat, Wo, bo, out);
}